// MyModel_5248450035921
// MI455X (gfx1250) — compile-verified
//
#include <hip/hip_runtime.h>
#include <hip/hip_bf16.h>
#include <math.h>

typedef float v2f __attribute__((ext_vector_type(2)));
typedef float v8f __attribute__((ext_vector_type(8)));

#define HIDN 32
#define NBMAX 1024

// ---------------------------------------------------------------------------
// Kernel 1: everything up to h_new, plus the two small 24-way heads.
// One block of 128 threads; all matrices are tiny (<=96x32).
// ---------------------------------------------------------------------------
struct SmallParams {
    const int* x; const float* hidden; const int* pos;
    const float* loc_emb; const float* arr_emb; const float* dur_emb; const float* pos_emb;
    const float* rW1[3]; const float* rb1[3]; const float* rW2[3]; const float* rb2[3];
    const float* W[3];   const float* b[3];
    const float* ln_g; const float* ln_b;
    const float* ode_W; const float* ode_b;
    const float* ode_rW1; const float* ode_rb1; const float* ode_rW2; const float* ode_rb2;
    const float* fc5_W; const float* fc5_b; const float* fc6_W; const float* fc6_b;
    const float* gru_Wih; const float* gru_bih; const float* gru_Whh; const float* gru_bhh;
    float* out; float* ws_h; int N;
};

__global__ void small_pipeline(SmallParams p) {
    const int t = threadIdx.x;                      // 128 threads
    __shared__ float xv[3][16], hsh[3][16], h2[3][16];
    __shared__ float point[15], dur5[5];
    __shared__ float odev[32], oh1[32], oh2[32], hvec[32], hnew[32], hp[32];
    __shared__ float gi[96], gh[96];
    __shared__ float a5[24], a6[24];
    __shared__ float stats[4];                      // mu, rsqrt(var), head5 sub, head6 sub

    // --- gather embeddings for the three chains ---
    if (t < 48) {
        int c = t >> 4, i = t & 15;
        const float* emb = (c == 0) ? p.loc_emb + (size_t)p.x[0] * 16
                         : (c == 1) ? p.arr_emb + (size_t)p.x[1] * 16
                                    : p.dur_emb + (size_t)p.x[2] * 16;
        xv[c][i] = emb[i];
    }
    __syncthreads();

    // --- two ResNet(16) blocks per chain ---
    for (int blk = 0; blk < 2; ++blk) {
        if (t < 48) { int c = t >> 4, i = t & 15; hsh[c][i] = fmaxf(xv[c][i], 0.f); }
        __syncthreads();
        if (t < 48) {
            int c = t >> 4, i = t & 15;
            const float* W1 = p.rW1[c] + blk * 256 + i * 16;
            float s = p.rb1[c][blk * 16 + i];
            for (int j = 0; j < 16; ++j) s += W1[j] * hsh[c][j];
            h2[c][i] = fmaxf(s, 0.f);
        }
        __syncthreads();
        if (t < 48) {
            int c = t >> 4, i = t & 15;
            const float* W2 = p.rW2[c] + blk * 256 + i * 16;
            float s = p.rb2[c][blk * 16 + i];
            for (int j = 0; j < 16; ++j) s += W2[j] * h2[c][j];
            xv[c][i] += 0.3f * s;
        }
        __syncthreads();
    }

    // --- final Linear(16->5) per chain; point = concat(loc, arr, dur) ---
    if (t < 15) {
        int c = t / 5, o = t % 5;
        const float* Wr = p.W[c] + o * 16;
        float s = p.b[c][o];
        for (int j = 0; j < 16; ++j) s += Wr[j] * xv[c][j];
        point[t] = s;
        if (c == 2) dur5[o] = s;                    // raw dur (pre-LayerNorm) feeds ODE
    }
    __syncthreads();

    // --- LayerNorm(15) ---
    if (t == 0) {
        float mu = 0.f;
        for (int i = 0; i < 15; ++i) mu += point[i];
        mu /= 15.f;
        float var = 0.f;
        for (int i = 0; i < 15; ++i) { float d = point[i] - mu; var += d * d; }
        var /= 15.f;
        stats[0] = mu; stats[1] = rsqrtf(var + 1e-5f);
    }
    __syncthreads();
    if (t < 15) point[t] = (point[t] - stats[0]) * stats[1] * p.ln_g[t] + p.ln_b[t];

    // --- ODE: Linear(5->32) ---
    if (t < 32) {
        const float* Wr = p.ode_W + t * 5;
        float s = p.ode_b[t];
        for (int j = 0; j < 5; ++j) s += Wr[j] * dur5[j];
        odev[t] = s;
        oh1[t] = fmaxf(s, 0.f);
    }
    __syncthreads();
    // --- ODE ResNet(32) ---
    if (t < 32) {
        const float* Wr = p.ode_rW1 + t * 32;
        float s = p.ode_rb1[t];
        for (int j = 0; j < 32; ++j) s += Wr[j] * oh1[j];
        oh2[t] = fmaxf(s, 0.f);
    }
    __syncthreads();
    if (t < 32) {
        const float* Wr = p.ode_rW2 + t * 32;
        float s = p.ode_rb2[t];
        for (int j = 0; j < 32; ++j) s += Wr[j] * oh2[j];
        hvec[t] = p.hidden[t] + (odev[t] + 0.3f * s);
    }
    __syncthreads();

    // --- GRU gates ---
    if (t < 96) {
        const float* Wi = p.gru_Wih + t * 15;
        float si = p.gru_bih[t];
        for (int j = 0; j < 15; ++j) si += Wi[j] * point[j];
        gi[t] = si;
        const float* Wh = p.gru_Whh + t * 32;
        float sh = p.gru_bhh[t];
        for (int j = 0; j < 32; ++j) sh += Wh[j] * hvec[j];
        gh[t] = sh;
    }
    __syncthreads();
    if (t < 32) {
        float r = 1.f / (1.f + expf(-(gi[t]      + gh[t])));
        float z = 1.f / (1.f + expf(-(gi[32 + t] + gh[32 + t])));
        float n = tanhf(gi[64 + t] + r * gh[64 + t]);
        float hn = (1.f - z) * n + z * hvec[t];
        hnew[t] = hn;
        p.ws_h[t] = hn;                               // for the big GEMV kernel
        p.out[(size_t)p.N + 48 + t] = hn;             // h_new output (after 24+24 heads)
        hp[t] = hn + p.pos_emb[(size_t)p.pos[0] * 32 + t];
    }
    __syncthreads();

    // --- fc5 / fc6 heads (24 x 32 each) ---
    if (t < 48) {
        int head = t / 24, o = t % 24;
        const float* Wr  = (head == 0 ? p.fc5_W : p.fc6_W) + o * 32;
        const float* bb  =  head == 0 ? p.fc5_b : p.fc6_b;
        const float* xin =  head == 0 ? hp : hnew;
        float s = bb[o];
        for (int j = 0; j < 32; ++j) s += Wr[j] * xin[j];
        (head == 0 ? a5 : a6)[o] = s;
    }
    __syncthreads();
    if (t < 2) {
        const float* a = (t == 0) ? a5 : a6;
        float m = a[0];
        for (int i = 1; i < 24; ++i) m = fmaxf(m, a[i]);
        float sum = 0.f;
        for (int i = 0; i < 24; ++i) sum += expf(a[i] - m);
        stats[2 + t] = m + logf(sum);
    }
    __syncthreads();
    if (t < 48) {
        int head = t / 24, o = t % 24;
        float v = (head == 0 ? a5[o] : a6[o]) - stats[2 + head];
        p.out[(size_t)p.N + head * 24 + o] = v;
    }
}

// ---------------------------------------------------------------------------
// Kernel 2: fc4 GEMV via V_WMMA_F32_16X16X4_F32.
// A (16x4, broadcast): A[m][k] = h[kc+k] for every m.
//   lane<16 holds {h[kc+0], h[kc+1]}, lane>=16 holds {h[kc+2], h[kc+3]}.
// B (4x16, weights):   B[k][n] = W[rowBase+n][kc+k].
//   lane loads float2 from W[(rowBase + (lane&15))*32 + kc + ((lane>>4)<<1)].
// D: every row identical; acc[0] at lane n (n<16) = dot(W[rowBase+n,:], h).
// Two 16-row tiles per iteration (two independent accumulation chains).
// No divergence in the tile loop -> EXEC all-1s at every WMMA.
// ---------------------------------------------------------------------------
__global__ void fc4_gemv_wmma(const float* __restrict__ W, const float* __restrict__ b,
                              const float* __restrict__ ws_h, float* __restrict__ out,
                              float* __restrict__ blockMax, int nRows) {
    const int lane   = threadIdx.x & 31;
    const int wave   = threadIdx.x >> 5;
    const int wpb    = blockDim.x >> 5;
    const int waveId = blockIdx.x * wpb + wave;
    const int nWaves = gridDim.x * wpb;

    const int col  = lane & 15;               // output row within a 16-row tile
    const int koff = (lane >> 4) << 1;        // 0 or 2 (k-split across wave halves)

    // Tile-invariant broadcast of h into A chunks.
    v2f av[8];
#pragma unroll
    for (int c = 0; c < 8; ++c) {
        v2f t;
        t.x = ws_h[c * 4 + koff + 0];
        t.y = ws_h[c * 4 + koff + 1];
        av[c] = t;
    }

    float localMax = -3.4e38f;
    const int nPairs = nRows >> 5;            // 32 rows per iteration

    for (int pair = waveId; pair < nPairs; pair += nWaves) {
        const int rowBase = pair << 5;
        const float* wp0 = W + (((size_t)(rowBase + col)) << 5) + koff;
        const float* wp1 = wp0 + (16 << 5);   // next 16-row tile
        v8f acc0 = {0.f, 0.f, 0.f, 0.f, 0.f, 0.f, 0.f, 0.f};
        v8f acc1 = {0.f, 0.f, 0.f, 0.f, 0.f, 0.f, 0.f, 0.f};
#pragma unroll
        for (int c = 0; c < 8; ++c) {
            v2f b0 = *(const v2f*)(wp0 + (c << 2));
            v2f b1 = *(const v2f*)(wp1 + (c << 2));
            acc0 = __builtin_amdgcn_wmma_f32_16x16x4_f32(
                false, av[c], false, b0, (short)0, acc0, false, false);
            acc1 = __builtin_amdgcn_wmma_f32_16x16x4_f32(
                false, av[c], false, b1, (short)0, acc1, false, false);
        }
        if (lane < 16) {                       // lanes 0..15 hold row 0 of D
            float v0 = acc0[0] + b[rowBase + col];
            float v1 = acc1[0] + b[rowBase + 16 + col];
            out[rowBase + col]      = v0;      // coalesced 64B store
            out[rowBase + 16 + col] = v1;
            localMax = fmaxf(localMax, fmaxf(v0, v1));
        }
    }

    __shared__ float smax[256];
    smax[threadIdx.x] = localMax;
    __syncthreads();
    for (int s = 128; s > 0; s >>= 1) {
        if (threadIdx.x < s) smax[threadIdx.x] = fmaxf(smax[threadIdx.x], smax[threadIdx.x + s]);
        __syncthreads();
    }
    if (threadIdx.x == 0) blockMax[blockIdx.x] = smax[0];
}

// ---------------------------------------------------------------------------
// log-softmax support kernels (deterministic fixed-order reductions)
// ---------------------------------------------------------------------------
__global__ void k_reduce_max(const float* __restrict__ bm, int n, float* __restrict__ gmax) {
    __shared__ float s[256];
    float m = -3.4e38f;
    for (int i = threadIdx.x; i < n; i += 256) m = fmaxf(m, bm[i]);
    s[threadIdx.x] = m;
    __syncthreads();
    for (int st = 128; st > 0; st >>= 1) {
        if (threadIdx.x < st) s[threadIdx.x] = fmaxf(s[threadIdx.x], s[threadIdx.x + st]);
        __syncthreads();
    }
    if (threadIdx.x == 0) *gmax = s[0];
}

__global__ void k_sumexp(const float* __restrict__ logits, int n,
                         const float* __restrict__ gmaxp, float* __restrict__ blockSum) {
    const float gmax = *gmaxp;
    float acc = 0.f;
    for (int i = blockIdx.x * blockDim.x + threadIdx.x; i < n; i += gridDim.x * blockDim.x)
        acc += expf(logits[i] - gmax);
    __shared__ float s[256];
    s[threadIdx.x] = acc;
    __syncthreads();
    for (int st = 128; st > 0; st >>= 1) {
        if (threadIdx.x < st) s[threadIdx.x] += s[threadIdx.x + st];
        __syncthreads();
    }
    if (threadIdx.x == 0) blockSum[blockIdx.x] = s[0];
}

__global__ void k_reduce_sum(const float* __restrict__ bs, int n,
                             const float* __restrict__ gmaxp, float* __restrict__ subp) {
    __shared__ float s[256];
    float acc = 0.f;
    for (int i = threadIdx.x; i < n; i += 256) acc += bs[i];
    s[threadIdx.x] = acc;
    __syncthreads();
    for (int st = 128; st > 0; st >>= 1) {
        if (threadIdx.x < st) s[threadIdx.x] += s[threadIdx.x + st];
        __syncthreads();
    }
    if (threadIdx.x == 0) *subp = *gmaxp + logf(s[0]);
}

__global__ void k_final(float* __restrict__ logits, int n, const float* __restrict__ subp) {
    const float sub = *subp;
    for (int i = blockIdx.x * blockDim.x + threadIdx.x; i < n; i += gridDim.x * blockDim.x)
        logits[i] -= sub;
}

// ---------------------------------------------------------------------------
extern "C" void kernel_launch(void* const* d_in, const int* in_sizes, int n_in,
                              void* d_out, int out_size, void* d_ws, size_t ws_size,
                              hipStream_t stream) {
    (void)n_in; (void)out_size;
    const int N = in_sizes[34];                 // fc4_b length = 1,000,000

    float* out = (float*)d_out;
    float* ws  = (float*)d_ws;

    // ws layout: [0..31] h_new | [32] gmax | [33] gmax+log(sum) | [64..) blockMax | then blockSum
    int NB = NBMAX;
    {
        long avail = ((long)(ws_size / sizeof(float)) - 64) / 2;
        if (avail < NB) NB = (avail > 0) ? (int)avail : 1;
    }
    float* ws_h     = ws;
    float* gmaxp    = ws + 32;
    float* subp     = ws + 33;
    float* blockMax = ws + 64;
    float* blockSum = ws + 64 + NB;

    SmallParams p;
    p.x       = (const int*)d_in[0];
    p.hidden  = (const float*)d_in[1];
    p.pos     = (const int*)d_in[2];
    p.loc_emb = (const float*)d_in[3];
    p.arr_emb = (const float*)d_in[4];
    p.dur_emb = (const float*)d_in[5];
    p.pos_emb = (const float*)d_in[6];
    for (int c = 0; c < 3; ++c) {
        int base = 7 + 6 * c;
        p.rW1[c] = (const float*)d_in[base + 0];
        p.rb1[c] = (const float*)d_in[base + 1];
        p.rW2[c] = (const float*)d_in[base + 2];
        p.rb2[c] = (const float*)d_in[base + 3];
        p.W[c]   = (const float*)d_in[base + 4];
        p.b[c]   = (const float*)d_in[base + 5];
    }
    p.ln_g    = (const float*)d_in[25];
    p.ln_b    = (const float*)d_in[26];
    p.ode_W   = (const float*)d_in[27];
    p.ode_b   = (const float*)d_in[28];
    p.ode_rW1 = (const float*)d_in[29];
    p.ode_rb1 = (const float*)d_in[30];
    p.ode_rW2 = (const float*)d_in[31];
    p.ode_rb2 = (const float*)d_in[32];
    const float* fc4_W = (const float*)d_in[33];
    const float* fc4_b = (const float*)d_in[34];
    p.fc5_W   = (const float*)d_in[35];
    p.fc5_b   = (const float*)d_in[36];
    p.fc6_W   = (const float*)d_in[37];
    p.fc6_b   = (const float*)d_in[38];
    p.gru_Wih = (const float*)d_in[39];
    p.gru_bih = (const float*)d_in[40];
    p.gru_Whh = (const float*)d_in[41];
    p.gru_bhh = (const float*)d_in[42];
    p.out  = out;
    p.ws_h = ws_h;
    p.N    = N;

    // 1) small pipeline -> h_new (+ small heads)
    small_pipeline<<<1, 128, 0, stream>>>(p);

    // 2) fc4 GEMV via WMMA + per-block max
    fc4_gemv_wmma<<<NB, 256, 0, stream>>>(fc4_W, fc4_b, ws_h, out, blockMax, N);

    // 3) global max
    k_reduce_max<<<1, 256, 0, stream>>>(blockMax, NB, gmaxp);

    // 4) per-block sum of exp(x - max)
    k_sumexp<<<NB, 256, 0, stream>>>(out, N, gmaxp, blockSum);

    // 5) total sum -> max + log(sum); then final subtraction
    k_reduce_sum<<<1, 256, 0, stream>>>(blockSum, NB, gmaxp, subp);
    k_final<<<NB, 256, 0, stream>>>(out, N, subp);
}